// Model_GIN_50869592655562
// MI455X (gfx1250) — compile-verified
//
#include <hip/hip_runtime.h>
#include <hip/hip_bf16.h>

#define N_NODES_MAX 100000
#define HID 64
#define DENSE 128
#define OUT_DIM 10
#define GIN_LAYERS 4
#define N_GRAPHS 128
#define BN_EPS 1e-5f

typedef float v2f __attribute__((ext_vector_type(2)));
typedef float v8f __attribute__((ext_vector_type(8)));

// ---------------------------------------------------------------------------
// Node embedding gather: x[n][:] = emb[x_idx[n]][:]
// ---------------------------------------------------------------------------
__global__ void embed_kernel(const int* __restrict__ idx,
                             const float* __restrict__ emb,
                             float* __restrict__ x, int n) {
  int g = blockIdx.x * blockDim.x + threadIdx.x;
  int node = g >> 4;          // 16 float4 per 64-float row
  int part = g & 15;
  if (node >= n) return;
  const float4* s = (const float4*)(emb + (size_t)idx[node] * HID);
  float4* d = (float4*)(x + (size_t)node * HID);
  d[part] = s[part];
}

// ---------------------------------------------------------------------------
// GIN aggregation: aggr[dst[e]] += x[src[e]]  (L2-resident scatter-add)
// ---------------------------------------------------------------------------
__global__ void scatter_kernel(const float* __restrict__ x,
                               const int* __restrict__ src,
                               const int* __restrict__ dst,
                               float* __restrict__ aggr, int nE) {
  int g = blockIdx.x * blockDim.x + threadIdx.x;
  int e = g >> 4;
  int part = g & 15;
  if (e >= nE) return;
  int s = src[e], d = dst[e];
  float4 v = ((const float4*)(x + (size_t)s * HID))[part];
  float* o = aggr + (size_t)d * HID + part * 4;
  atomicAdd(o + 0, v.x);
  atomicAdd(o + 1, v.y);
  atomicAdd(o + 2, v.z);
  atomicAdd(o + 3, v.w);
}

// ---------------------------------------------------------------------------
// GEMM1: z1[N,128] = ((1+eps)*x + aggr)[N,64] @ W[64,128]
// block = 256 thr = 8 waves, 128-row strip, f32 WMMA 16x16x4.
// ---------------------------------------------------------------------------
#define LDA 68   // padded LDS row stride (68 % 64 == 4 -> conflict-free)

__global__ __launch_bounds__(256) void gemm1_kernel(
    const float* __restrict__ x, const float* __restrict__ aggr,
    const float* __restrict__ epsArr, int layer,
    const float* __restrict__ W,    // [64,128]
    float* __restrict__ z, int nrows) {
  __shared__ float As[128 * LDA];
  const float epsv = 1.0f + epsArr[layer];
  const int rowBase = blockIdx.x * 128;
  const int tid = threadIdx.x;

  // Stage A tile: h = (1+eps)*x + aggr (128 rows x 64 cols), coalesced float4
  {
    int cpart = (tid & 15) * 4;
    int r0 = tid >> 4;
#pragma unroll
    for (int i = 0; i < 8; ++i) {
      int r = r0 + 16 * i;
      int gr = rowBase + r;
      int grc = gr < nrows ? gr : (nrows - 1);   // clamp; OOB rows discarded
      float4 xv = *(const float4*)(x + (size_t)grc * HID + cpart);
      float4 av = *(const float4*)(aggr + (size_t)grc * HID + cpart);
      float* s = &As[r * LDA + cpart];
      s[0] = epsv * xv.x + av.x;
      s[1] = epsv * xv.y + av.y;
      s[2] = epsv * xv.z + av.z;
      s[3] = epsv * xv.w + av.w;
    }
  }
  __syncthreads();

  const int wave = tid >> 5;           // wave32
  const int lane = tid & 31;
  const int m0 = wave * 16;
  const int ln = lane & 15;
  const int khalf = (lane >> 4) * 2;   // lanes 0-15 -> K pair {0,1}; 16-31 -> {2,3}

  v8f acc[8] = {};
#pragma unroll
  for (int k0 = 0; k0 < 64; k0 += 4) {
    int kk = k0 + khalf;
    v2f a = *(const v2f*)(&As[(m0 + ln) * LDA + kk]);   // ds_load_b64
#pragma unroll
    for (int t = 0; t < 8; ++t) {
      v2f b;
      b.x = W[kk * DENSE + t * 16 + ln];
      b.y = W[(kk + 1) * DENSE + t * 16 + ln];
      acc[t] = __builtin_amdgcn_wmma_f32_16x16x4_f32(
          false, a, false, b, (short)0, acc[t], false, false);
    }
  }

  int rowHalf = (lane >> 4) * 8;
#pragma unroll
  for (int r = 0; r < 8; ++r) {
    int gr = rowBase + m0 + rowHalf + r;
    if (gr < nrows) {
#pragma unroll
      for (int t = 0; t < 8; ++t)
        z[(size_t)gr * DENSE + t * 16 + ln] = acc[t][r];
    }
  }
}

// ---------------------------------------------------------------------------
// GEMM2: z2[N,64] = relu(bn(z1))[N,128] @ W[128,64]
// BN affine + ReLU fused into A-tile staging; K split in two LDS phases.
// ---------------------------------------------------------------------------
__global__ __launch_bounds__(256) void gemm2_kernel(
    const float* __restrict__ z1,
    const float* __restrict__ bnp,   // [0..127] scale, [128..255] shift
    const float* __restrict__ W,     // [128,64]
    float* __restrict__ z2, int nrows) {
  __shared__ float As[128 * LDA];
  const int rowBase = blockIdx.x * 128;
  const int tid = threadIdx.x;
  const int wave = tid >> 5;
  const int lane = tid & 31;
  const int m0 = wave * 16;
  const int ln = lane & 15;
  const int khalf = (lane >> 4) * 2;

  v8f acc[4] = {};
  for (int phase = 0; phase < 2; ++phase) {
    const int kbase = phase * 64;
    __syncthreads();
    {
      int cpart = (tid & 15) * 4;
      int r0 = tid >> 4;
#pragma unroll
      for (int i = 0; i < 8; ++i) {
        int r = r0 + 16 * i;
        int gr = rowBase + r;
        int grc = gr < nrows ? gr : (nrows - 1);
        float4 v = *(const float4*)(z1 + (size_t)grc * DENSE + kbase + cpart);
        float* s = &As[r * LDA + cpart];
        int c = kbase + cpart;
        float h0 = v.x * bnp[c + 0] + bnp[DENSE + c + 0];
        float h1 = v.y * bnp[c + 1] + bnp[DENSE + c + 1];
        float h2 = v.z * bnp[c + 2] + bnp[DENSE + c + 2];
        float h3 = v.w * bnp[c + 3] + bnp[DENSE + c + 3];
        s[0] = h0 > 0.f ? h0 : 0.f;
        s[1] = h1 > 0.f ? h1 : 0.f;
        s[2] = h2 > 0.f ? h2 : 0.f;
        s[3] = h3 > 0.f ? h3 : 0.f;
      }
    }
    __syncthreads();
#pragma unroll
    for (int k0 = 0; k0 < 64; k0 += 4) {
      int kk = k0 + khalf;
      v2f a = *(const v2f*)(&As[(m0 + ln) * LDA + kk]);
      int kw = kbase + kk;
#pragma unroll
      for (int t = 0; t < 4; ++t) {
        v2f b;
        b.x = W[kw * HID + t * 16 + ln];
        b.y = W[(kw + 1) * HID + t * 16 + ln];
        acc[t] = __builtin_amdgcn_wmma_f32_16x16x4_f32(
            false, a, false, b, (short)0, acc[t], false, false);
      }
    }
  }

  int rowHalf = (lane >> 4) * 8;
#pragma unroll
  for (int r = 0; r < 8; ++r) {
    int gr = rowBase + m0 + rowHalf + r;
    if (gr < nrows) {
#pragma unroll
      for (int t = 0; t < 4; ++t)
        z2[(size_t)gr * HID + t * 16 + ln] = acc[t][r];
    }
  }
}

// ---------------------------------------------------------------------------
// Per-column sum / sum-of-squares over the batch dim (grid-stride + atomics)
// ---------------------------------------------------------------------------
__global__ void colstats_kernel(const float* __restrict__ z, int nrows, int C,
                                float* __restrict__ stats) {
  int tid = threadIdx.x;
  int col = tid % C;
  int rOff = tid / C;
  int tpc = 256 / C;
  float s = 0.f, sq = 0.f;
  for (int r = blockIdx.x * tpc + rOff; r < nrows; r += gridDim.x * tpc) {
    float v = z[(size_t)r * C + col];
    s += v;
    sq += v * v;
  }
  __shared__ float sh[256], shq[256];
  sh[tid] = s;
  shq[tid] = sq;
  __syncthreads();
  if (rOff == 0) {
    for (int i = 1; i < tpc; ++i) {
      s += sh[tid + i * C];
      sq += shq[tid + i * C];
    }
    atomicAdd(&stats[col], s);
    atomicAdd(&stats[C + col], sq);
  }
}

// stats -> (scale, shift) so bn(x) = x*scale + shift
__global__ void bnfinal_kernel(const float* __restrict__ stats,
                               const float* __restrict__ g,
                               const float* __restrict__ b, float invN, int C,
                               float* __restrict__ bnp) {
  int c = blockIdx.x * blockDim.x + threadIdx.x;
  if (c >= C) return;
  float m = stats[c] * invN;
  float v = stats[C + c] * invN - m * m;
  float inv = rsqrtf(v + BN_EPS);
  float sc = g[c] * inv;
  bnp[c] = sc;
  bnp[C + c] = b[c] - m * sc;
}

__global__ void apply_bn_relu_kernel(const float* __restrict__ z,
                                     const float* __restrict__ bnp,
                                     float* __restrict__ out, int n, int C) {
  int g = blockIdx.x * blockDim.x + threadIdx.x;
  if (g >= n * C) return;
  int c = g % C;
  float v = z[g] * bnp[c] + bnp[C + c];
  out[g] = v > 0.f ? v : 0.f;
}

// ---------------------------------------------------------------------------
// Global mean pool (sums + counts via atomics)
// ---------------------------------------------------------------------------
__global__ void pool_kernel(const float* __restrict__ x,
                            const int* __restrict__ batch,
                            float* __restrict__ sums,
                            float* __restrict__ counts, int n) {
  int g = blockIdx.x * blockDim.x + threadIdx.x;
  int node = g >> 4;
  int part = g & 15;
  if (node >= n) return;
  int b = batch[node];
  float4 v = ((const float4*)(x + (size_t)node * HID))[part];
  float* o = sums + (size_t)b * HID + part * 4;
  atomicAdd(o + 0, v.x);
  atomicAdd(o + 1, v.y);
  atomicAdd(o + 2, v.z);
  atomicAdd(o + 3, v.w);
  if (part == 0) atomicAdd(&counts[b], 1.0f);
}

__global__ void reps_kernel(const float* __restrict__ sums,
                            const float* __restrict__ counts,
                            float* __restrict__ reps) {
  int g = blockIdx.x * blockDim.x + threadIdx.x;
  if (g >= N_GRAPHS * HID) return;
  float c = counts[g / HID];
  c = c > 1.f ? c : 1.f;
  reps[g] = sums[g] / c;
}

// ---------------------------------------------------------------------------
// Tiny 128-row MLP head (negligible FLOPs; plain kernels)
// ---------------------------------------------------------------------------
__global__ void small_gemm_kernel(const float* __restrict__ A,
                                  const float* __restrict__ B,
                                  float* __restrict__ C, int M, int K, int N) {
  int g = blockIdx.x * blockDim.x + threadIdx.x;
  if (g >= M * N) return;
  int m = g / N, n = g % N;
  float acc = 0.f;
  for (int k = 0; k < K; ++k) acc += A[m * K + k] * B[k * N + n];
  C[g] = acc;
}

__global__ void bn_rows_kernel(float* __restrict__ Z, const float* __restrict__ g,
                               const float* __restrict__ b, int M, int C) {
  int c = threadIdx.x;
  if (c >= C) return;
  float s = 0.f, sq = 0.f;
  for (int m = 0; m < M; ++m) {
    float v = Z[m * C + c];
    s += v;
    sq += v * v;
  }
  float mean = s / M;
  float var = sq / M - mean * mean;
  float inv = rsqrtf(var + BN_EPS);
  float sc = g[c] * inv, sh = b[c] - mean * sc;
  for (int m = 0; m < M; ++m) {
    float v = Z[m * C + c] * sc + sh;
    Z[m * C + c] = v > 0.f ? v : 0.f;
  }
}

__global__ void head_out_kernel(const float* __restrict__ H,
                                const float* __restrict__ Wlin,
                                const float* __restrict__ blin,
                                float* __restrict__ out) {
  int g = blockIdx.x * blockDim.x + threadIdx.x;
  if (g >= N_GRAPHS * OUT_DIM) return;
  int m = g / OUT_DIM, o = g % OUT_DIM;
  float acc = blin[o];
  for (int k = 0; k < DENSE; ++k) acc += H[m * DENSE + k] * Wlin[k * OUT_DIM + o];
  out[g] = acc;
}

// ---------------------------------------------------------------------------
extern "C" void kernel_launch(void* const* d_in, const int* in_sizes, int n_in,
                              void* d_out, int out_size, void* d_ws,
                              size_t ws_size, hipStream_t stream) {
  const int* x_idx = (const int*)d_in[0];
  const int* edge = (const int*)d_in[1];
  const int* batch = (const int*)d_in[2];
  const float* emb = (const float*)d_in[3];
  const float* eps = (const float*)d_in[4];
  const float* W1 = (const float*)d_in[5];
  const float* g1 = (const float*)d_in[6];
  const float* b1 = (const float*)d_in[7];
  const float* W2 = (const float*)d_in[8];
  const float* g2 = (const float*)d_in[9];
  const float* b2 = (const float*)d_in[10];
  const float* Wf1 = (const float*)d_in[11];
  const float* gf1 = (const float*)d_in[12];
  const float* bf1 = (const float*)d_in[13];
  const float* Wf2 = (const float*)d_in[14];
  const float* gf2 = (const float*)d_in[15];
  const float* bf2 = (const float*)d_in[16];
  const float* Wlin = (const float*)d_in[17];
  const float* blin = (const float*)d_in[18];
  float* out = (float*)d_out;

  const int nN = in_sizes[0];
  const int nE = in_sizes[1] / 2;

  // bump-allocate workspace
  char* ws = (char*)d_ws;
  size_t off = 0;
  auto alloc = [&](size_t nbytes) -> void* {
    void* p = ws + off;
    off += (nbytes + 255) & ~(size_t)255;
    return p;
  };
  float* x = (float*)alloc((size_t)nN * HID * 4);
  float* aggr = (float*)alloc((size_t)nN * HID * 4);  // reused as z2
  float* z1 = (float*)alloc((size_t)nN * DENSE * 4);
  float* stats = (float*)alloc(2 * DENSE * 4);
  float* bnp1 = (float*)alloc(2 * DENSE * 4);
  float* bnp2 = (float*)alloc(2 * HID * 4);
  float* sums = (float*)alloc((size_t)N_GRAPHS * HID * 4);
  float* counts = (float*)alloc(N_GRAPHS * 4);
  float* reps = (float*)alloc((size_t)N_GRAPHS * HID * 4);
  float* h1 = (float*)alloc((size_t)N_GRAPHS * DENSE * 4);
  float* h2 = (float*)alloc((size_t)N_GRAPHS * DENSE * 4);
  float* z2 = aggr;  // aggr dead after gemm1 -> reuse as gemm2 output

  const int gemmBlocks = (nN + 127) / 128;

  embed_kernel<<<(nN * 16 + 255) / 256, 256, 0, stream>>>(x_idx, emb, x, nN);

  for (int l = 0; l < GIN_LAYERS; ++l) {
    hipMemsetAsync(aggr, 0, (size_t)nN * HID * 4, stream);
    scatter_kernel<<<(nE * 16 + 255) / 256, 256, 0, stream>>>(
        x, edge, edge + nE, aggr, nE);

    gemm1_kernel<<<gemmBlocks, 256, 0, stream>>>(
        x, aggr, eps, l, W1 + (size_t)l * HID * DENSE, z1, nN);

    hipMemsetAsync(stats, 0, 2 * DENSE * 4, stream);
    colstats_kernel<<<512, 256, 0, stream>>>(z1, nN, DENSE, stats);
    bnfinal_kernel<<<1, DENSE, 0, stream>>>(stats, g1 + l * DENSE,
                                            b1 + l * DENSE, 1.0f / nN, DENSE,
                                            bnp1);

    gemm2_kernel<<<gemmBlocks, 256, 0, stream>>>(
        z1, bnp1, W2 + (size_t)l * DENSE * HID, z2, nN);

    hipMemsetAsync(stats, 0, 2 * HID * 4, stream);
    colstats_kernel<<<512, 256, 0, stream>>>(z2, nN, HID, stats);
    bnfinal_kernel<<<1, HID, 0, stream>>>(stats, g2 + l * HID, b2 + l * HID,
                                          1.0f / nN, HID, bnp2);

    apply_bn_relu_kernel<<<(nN * HID + 255) / 256, 256, 0, stream>>>(
        z2, bnp2, x, nN, HID);
  }

  hipMemsetAsync(sums, 0, (size_t)N_GRAPHS * HID * 4, stream);
  hipMemsetAsync(counts, 0, N_GRAPHS * 4, stream);
  pool_kernel<<<(nN * 16 + 255) / 256, 256, 0, stream>>>(x, batch, sums,
                                                         counts, nN);
  reps_kernel<<<(N_GRAPHS * HID + 255) / 256, 256, 0, stream>>>(sums, counts,
                                                                reps);
  small_gemm_kernel<<<(N_GRAPHS * DENSE + 255) / 256, 256, 0, stream>>>(
      reps, Wf1, h1, N_GRAPHS, HID, DENSE);
  bn_rows_kernel<<<1, DENSE, 0, stream>>>(h1, gf1, bf1, N_GRAPHS, DENSE);
  small_gemm_kernel<<<(N_GRAPHS * DENSE + 255) / 256, 256, 0, stream>>>(
      h1, Wf2, h2, N_GRAPHS, DENSE, DENSE);
  bn_rows_kernel<<<1, DENSE, 0, stream>>>(h2, gf2, bf2, N_GRAPHS, DENSE);
  head_out_kernel<<<(N_GRAPHS * OUT_DIM + 255) / 256, 256, 0, stream>>>(
      h2, Wlin, blin, out);
}